// CausalSelfAttentionLast_30880814858987
// MI455X (gfx1250) — compile-verified
//
#include <hip/hip_runtime.h>
#include <hip/hip_bf16.h>

#define B_ 4
#define T_ 4096
#define C_ 2048
#define NCHUNK 32
#define SCALE_ 0.022097086912079612f   // 1/sqrt(2048)

typedef __attribute__((ext_vector_type(2))) float v2f;
typedef __attribute__((ext_vector_type(8))) float v8f;

// ---------------------------------------------------------------------------
// D[16,C] = A[16,C_k] @ W[C,C] + bias, where A has 4 valid rows (rest zero).
// One wave per 16-column tile of the output (C/16 = 128 waves total).
// Uses V_WMMA_F32_16X16X4_F32 (fp32 in, fp32 out, exact vs reference).
//   A 16x4 layout: lane&15 = M row; lane<16 holds K=k0,k0+1; lane>=16 holds K=k0+2,k0+3.
//   B 4x16 layout: lane&15 = N col; same K split across lane halves.
//   D layout: lanes 0-15: vgpr r = D[r][lane]; lanes 16-31: vgpr r = D[r+8][lane-16].
// ---------------------------------------------------------------------------
__global__ void rowgemm16_kernel(const float* __restrict__ X, long long xstride,
                                 const float* __restrict__ W,
                                 const float* __restrict__ bias,
                                 float* __restrict__ out) {
  int wave = (blockIdx.x * blockDim.x + threadIdx.x) >> 5;   // 0..127
  int lane = threadIdx.x & 31;
  int l15  = lane & 15;
  int kh   = lane >> 4;            // 0 or 1 -> K half
  int n0   = wave * 16;

  const float* xrow = X + (long long)(l15 & 3) * xstride;    // row m (replicated mod 4)
  float msel = (l15 < 4) ? 1.0f : 0.0f;                      // zero-pad rows 4..15

  v8f acc = {};
#pragma unroll 4
  for (int k0 = 0; k0 < C_; k0 += 4) {
    int k = k0 + 2 * kh;
    v2f a, b;
    a.x = xrow[k]     * msel;
    a.y = xrow[k + 1] * msel;
    b.x = W[(long long)k       * C_ + n0 + l15];
    b.y = W[(long long)(k + 1) * C_ + n0 + l15];
    acc = __builtin_amdgcn_wmma_f32_16x16x4_f32(false, a, false, b,
                                                (short)0, acc, false, false);
  }
  if (lane < 16) {                 // rows 0..3 live in acc[0..3] of lanes 0..15
    int n = n0 + lane;
    float bv = bias[n];
#pragma unroll
    for (int r = 0; r < 4; ++r)
      out[r * C_ + n] = acc[r] + bv;
  }
}

// ---------------------------------------------------------------------------
// KQ[b][c] = sum_j Wk[c][j] * Q[b][j]   (M=C rows of Wk, N=4 cols padded to 16)
// One wave per 16-row tile of c (128 waves).
// ---------------------------------------------------------------------------
__global__ void leftgemm_kernel(const float* __restrict__ W,   // Wk [C,C]
                                const float* __restrict__ Q,   // [4,C]
                                float* __restrict__ KQ) {      // [4,C]
  int wave = (blockIdx.x * blockDim.x + threadIdx.x) >> 5;
  int lane = threadIdx.x & 31;
  int l15  = lane & 15;
  int kh   = lane >> 4;
  int mt   = wave * 16;                                        // c base

  const float* wrow = W + (long long)(mt + l15) * C_;          // A row = Wk[c][*]
  const float* qrow = Q + (long long)(l15 & 3) * C_;           // B col n = Q[n][*]
  float nsel = (l15 < 4) ? 1.0f : 0.0f;

  v8f acc = {};
#pragma unroll 4
  for (int k0 = 0; k0 < C_; k0 += 4) {
    int k = k0 + 2 * kh;
    v2f a, b;
    a.x = wrow[k];
    a.y = wrow[k + 1];
    b.x = qrow[k]     * nsel;
    b.y = qrow[k + 1] * nsel;
    acc = __builtin_amdgcn_wmma_f32_16x16x4_f32(false, a, false, b,
                                                (short)0, acc, false, false);
  }
  // D[M=c][N=b]; only b = l15 < 4 is valid
  if (l15 < 4) {
    int cbase = mt + ((lane < 16) ? 0 : 8);
#pragma unroll
    for (int r = 0; r < 8; ++r)
      KQ[l15 * C_ + cbase + r] = acc[r];
  }
}

// qdotbk[b] = dot(q[b], bk)  (tiny; bk is typically zero but handled generally)
__global__ void qdotbk_kernel(const float* __restrict__ q,
                              const float* __restrict__ bk,
                              float* __restrict__ outv) {
  int lane = threadIdx.x & 31;
  int b    = threadIdx.x >> 5;            // 4 waves, one per batch
  float s = 0.f;
  for (int j = lane; j < C_; j += 32) s += q[b * C_ + j] * bk[j];
  for (int off = 16; off; off >>= 1) s += __shfl_down(s, off);
  if (lane == 0) outv[b] = s;
}

// ---------------------------------------------------------------------------
// logits[b][t] = (x[b,t,:] . kq[b] + qdotbk[b]) * scale     (pass 1 over x)
// One wave per (b,t); 16384 waves, b128 streams (512B per wave-instr).
// ---------------------------------------------------------------------------
__global__ void logits_kernel(const float* __restrict__ x,
                              const float* __restrict__ kq,
                              const float* __restrict__ qdotbk,
                              float* __restrict__ wei) {
  int gw   = (blockIdx.x * blockDim.x + threadIdx.x) >> 5;    // 0..B*T-1
  int lane = threadIdx.x & 31;
  int b    = gw >> 12;                                        // /T_
  const float* xr  = x  + (long long)gw * C_;                 // gw = b*T + t
  const float* kqb = kq + b * C_;
  float s = 0.f;
#pragma unroll 4
  for (int j = lane * 4; j < C_; j += 128) {
    float4 xv = *reinterpret_cast<const float4*>(xr + j);
    float4 kv = *reinterpret_cast<const float4*>(kqb + j);
    s += xv.x * kv.x + xv.y * kv.y + xv.z * kv.z + xv.w * kv.w;
  }
  for (int off = 16; off; off >>= 1) s += __shfl_down(s, off);
  if (lane == 0) wei[gw] = (s + qdotbk[b]) * SCALE_;
}

// ---------------------------------------------------------------------------
// In-place softmax over T per batch (one block per b).
// ---------------------------------------------------------------------------
__global__ void softmax_kernel(float* __restrict__ wei) {
  float* w = wei + (long long)blockIdx.x * T_;
  __shared__ float sdata[8];
  __shared__ float sbc[2];
  int tid  = threadIdx.x;                 // 0..255
  int lane = tid & 31, wv = tid >> 5;

  float mx = -3.4e38f;
  for (int t = tid; t < T_; t += 256) mx = fmaxf(mx, w[t]);
  for (int off = 16; off; off >>= 1) mx = fmaxf(mx, __shfl_down(mx, off));
  if (lane == 0) sdata[wv] = mx;
  __syncthreads();
  if (tid == 0) {
    float m = sdata[0];
    for (int i = 1; i < 8; ++i) m = fmaxf(m, sdata[i]);
    sbc[0] = m;
  }
  __syncthreads();
  mx = sbc[0];

  float sum = 0.f;
  for (int t = tid; t < T_; t += 256) {
    float e = expf(w[t] - mx);
    w[t] = e;
    sum += e;
  }
  __syncthreads();
  for (int off = 16; off; off >>= 1) sum += __shfl_down(sum, off);
  if (lane == 0) sdata[wv] = sum;
  __syncthreads();
  if (tid == 0) {
    float s = 0.f;
    for (int i = 0; i < 8; ++i) s += sdata[i];
    sbc[1] = 1.0f / s;
  }
  __syncthreads();
  float inv = sbc[1];
  for (int t = tid; t < T_; t += 256) w[t] *= inv;
}

// ---------------------------------------------------------------------------
// partial[b][chunk][c..c+3] = sum over (T/NCHUNK) t's of wei[b][t]*x[b][t][c..c+3]
// Pass 2 over x. Each thread owns 4 consecutive channels -> global_load_b128
// (512B per wave-instruction), maximizing HBM/L2 bandwidth per instruction.
// Grid = B * (C/1024) * NCHUNK = 256 blocks of 256 threads.
// ---------------------------------------------------------------------------
__global__ void wsum_partial_kernel(const float* __restrict__ x,
                                    const float* __restrict__ wei,
                                    float* __restrict__ partial) {
  const int CT = C_ / 1024;               // 2 column-tiles of 1024 channels
  int idx   = blockIdx.x;
  int chunk = idx % NCHUNK; idx /= NCHUNK;
  int ct    = idx % CT;     idx /= CT;
  int b     = idx;
  int c     = ct * 1024 + threadIdx.x * 4;
  int t0    = chunk * (T_ / NCHUNK);      // 128 rows per chunk
  const float* xb = x   + ((long long)b * T_ + t0) * C_ + c;
  const float* wb = wei + b * T_ + t0;
  float4 acc = {0.f, 0.f, 0.f, 0.f};
#pragma unroll 4
  for (int t = 0; t < T_ / NCHUNK; ++t) {
    float  w  = wb[t];
    float4 xv = *reinterpret_cast<const float4*>(xb + (long long)t * C_);
    acc.x += w * xv.x;
    acc.y += w * xv.y;
    acc.z += w * xv.z;
    acc.w += w * xv.w;
  }
  *reinterpret_cast<float4*>(partial + (long long)(b * NCHUNK + chunk) * C_ + c) = acc;
}

__global__ void wsum_reduce_kernel(const float* __restrict__ partial,
                                   float* __restrict__ xbv) {
  int i = (blockIdx.x * 256 + threadIdx.x) * 4;   // over B*C = 8192 floats
  int b = i / C_, c = i % C_;
  float4 s = {0.f, 0.f, 0.f, 0.f};
#pragma unroll
  for (int ch = 0; ch < NCHUNK; ++ch) {
    float4 p = *reinterpret_cast<const float4*>(
        partial + (long long)(b * NCHUNK + ch) * C_ + c);
    s.x += p.x; s.y += p.y; s.z += p.z; s.w += p.w;
  }
  *reinterpret_cast<float4*>(xbv + i) = s;
}

// ---------------------------------------------------------------------------
extern "C" void kernel_launch(void* const* d_in, const int* in_sizes, int n_in,
                              void* d_out, int out_size, void* d_ws, size_t ws_size,
                              hipStream_t stream) {
  (void)in_sizes; (void)n_in; (void)out_size; (void)ws_size;
  const float* x  = (const float*)d_in[0];
  const float* Wk = (const float*)d_in[1];
  const float* bk = (const float*)d_in[2];
  const float* Wq = (const float*)d_in[3];
  const float* bq = (const float*)d_in[4];
  const float* Wv = (const float*)d_in[5];
  const float* bv = (const float*)d_in[6];
  const float* Wp = (const float*)d_in[7];
  const float* bp = (const float*)d_in[8];

  float* out = (float*)d_out;              // [B,C] out_last
  float* wei = out + B_ * C_;              // [B,T] attention weights (output #2)

  float* ws      = (float*)d_ws;
  float* q       = ws;                     // [4,C]
  float* kq      = ws + 8192;              // [4,C]
  float* xbv     = ws + 16384;             // [4,C] weighted x rows
  float* vv      = ws + 24576;             // [4,C] (xbv@Wv+bv)
  float* qbk     = ws + 32768;             // [4]
  float* partial = ws + 32800;             // [4,NCHUNK,C] = 1MB

  // 1. q[b] = x[b,-1,:] @ Wq + bq          (WMMA fp32)
  rowgemm16_kernel<<<16, 256, 0, stream>>>(x + (long long)(T_ - 1) * C_,
                                           (long long)T_ * C_, Wq, bq, q);
  // 2. kq[b] = Wk @ q[b]; qbk[b] = q[b].bk  (WMMA fp32)
  leftgemm_kernel<<<16, 256, 0, stream>>>(Wk, q, kq);
  qdotbk_kernel<<<1, 128, 0, stream>>>(q, bk, qbk);
  // 3. logits over x (pass 1, HBM-bound)
  logits_kernel<<<(B_ * T_) / 8, 256, 0, stream>>>(x, kq, qbk, wei);
  // 4. softmax per batch -> final wei output
  softmax_kernel<<<B_, 256, 0, stream>>>(wei);
  // 5. xbv[b] = sum_t wei[b,t] * x[b,t,:]  (pass 2, L2-hot, b128 streams)
  wsum_partial_kernel<<<B_ * (C_ / 1024) * NCHUNK, 256, 0, stream>>>(x, wei, partial);
  wsum_reduce_kernel<<<(B_ * C_) / 1024, 256, 0, stream>>>(partial, xbv);
  // 6. vv = xbv @ Wv + bv                  (WMMA fp32)
  rowgemm16_kernel<<<16, 256, 0, stream>>>(xbv, (long long)C_, Wv, bv, vv);
  // 7. out_last = vv @ Wp + bp             (WMMA fp32)
  rowgemm16_kernel<<<16, 256, 0, stream>>>(vv, (long long)C_, Wp, bp, out);
}